// GNN_67396626809445
// MI455X (gfx1250) — compile-verified
//
#include <hip/hip_runtime.h>

typedef __attribute__((ext_vector_type(2))) float v2f;
typedef __attribute__((ext_vector_type(8))) float v8f;

#define WPB 8                      // waves per block (wave32)
#define GEMM_THREADS (WPB * 32)
#define GEMM_ROWS (WPB * 16)       // rows per block (16 per wave)
#define EPSV 1e-5f

// ---------------------------------------------------------------------------
// Generic C = A[M,K] @ B[K,32] (+bias), K <= 32, with optional fused
// attention-dot  e[m,h] = sum_f C[m, h*16+f] * attn[h*16+f]  (H=2, F=16).
// Uses V_WMMA_F32_16X16X4_F32: A tile 16x4, B tile 4x16, C 16x16 (v8f acc).
// A-layout: lanes 0-15 row M=lane, VGPR0/1 = K0/K1; lanes 16-31 = K2/K3.
// C-layout: VGPR i -> row i (lanes 0-15) / row 8+i (lanes 16-31), col = lane&15.
// ---------------------------------------------------------------------------
__global__ void gemm32_wmma(const float* __restrict__ A, int M, int K,
                            const float* __restrict__ B,      // K x 32
                            const float* __restrict__ bias,   // 32 or null
                            float* __restrict__ C,            // M x 32 or null
                            const float* __restrict__ attn,   // 32 or null
                            float* __restrict__ e_out)        // M x 2 or null
{
    __shared__ float ldsB[32][33];
    __shared__ float ldsA[WPB][16][33];

    const int tid = threadIdx.x;
    // Stage B, zero-padded to 32 rows of K.
    for (int i = tid; i < 32 * 32; i += GEMM_THREADS) {
        int k = i >> 5, n = i & 31;
        ldsB[k][n] = (k < K) ? B[k * 32 + n] : 0.0f;
    }
    const int wid  = tid >> 5;
    const int lane = tid & 31;
    const int half = lane >> 4;
    const int l15  = lane & 15;
    const int row0 = blockIdx.x * GEMM_ROWS + wid * 16;

    // Stage this wave's 16xK A tile (zero padded to 16x32), coalesced in k.
    for (int i = lane; i < 16 * 32; i += 32) {
        int r = i >> 5, k = i & 31;
        int row = row0 + r;
        ldsA[wid][r][k] = (row < M && k < K) ? A[(long)row * K + k] : 0.0f;
    }
    __syncthreads();

    for (int nt = 0; nt < 2; ++nt) {          // two 16-col output tiles (=2 heads)
        v8f c = {};
#pragma unroll
        for (int kb = 0; kb < 8; ++kb) {      // K padded to 32, steps of 4
            const int k0 = kb * 4 + half * 2;
            v2f a, b;
            a.x = ldsA[wid][l15][k0];
            a.y = ldsA[wid][l15][k0 + 1];
            b.x = ldsB[k0][nt * 16 + l15];
            b.y = ldsB[k0 + 1][nt * 16 + l15];
            c = __builtin_amdgcn_wmma_f32_16x16x4_f32(false, a, false, b,
                                                      (short)0, c, false, false);
        }
        const int col = nt * 16 + l15;
        if (C) {
            const float bb = bias ? bias[col] : 0.0f;
#pragma unroll
            for (int i = 0; i < 8; ++i) {
                int row = row0 + i + half * 8;
                if (row < M) C[(long)row * 32 + col] = c[i] + bb;
            }
        }
        if (attn && e_out) {
            const float av = attn[col];
#pragma unroll
            for (int i = 0; i < 8; ++i) {
                float p = c[i] * av;
                p += __shfl_xor(p, 1);
                p += __shfl_xor(p, 2);
                p += __shfl_xor(p, 4);
                p += __shfl_xor(p, 8);        // sum over 16 lanes of this half
                int row = row0 + i + half * 8;
                if (l15 == 0 && row < M) e_out[row * 2 + nt] = p;
            }
        }
    }
}

// ----------------------- elementwise / init helpers ------------------------
__global__ void fill_f32(float* p, int n, float v) {
    int t = blockIdx.x * blockDim.x + threadIdx.x;
    if (t < n) p[t] = v;
}
__global__ void fill_u32(unsigned* p, int n, unsigned v) {
    int t = blockIdx.x * blockDim.x + threadIdx.x;
    if (t < n) p[t] = v;
}
// out[n,32] = b0 + (b1?) + (b2?)  broadcast per column
__global__ void init_bias3(float* out, int n_rows, const float* b0,
                           const float* b1, const float* b2) {
    int t = blockIdx.x * blockDim.x + threadIdx.x;
    if (t >= n_rows * 32) return;
    int c = t & 31;
    float v = b0[c];
    if (b1) v += b1[c];
    if (b2) v += b2[c];
    out[t] = v;
}
__global__ void scale_inplace(float* p, int n, float s) {
    int t = blockIdx.x * blockDim.x + threadIdx.x;
    if (t < n) p[t] *= s;
}

// ------------------------- edge softmax kernels ----------------------------
__device__ __forceinline__ unsigned fenc(float f) {
    unsigned u = __float_as_uint(f);
    return (u & 0x80000000u) ? ~u : (u | 0x80000000u);   // order-preserving
}
__device__ __forceinline__ float fdec(unsigned e) {
    unsigned u = (e & 0x80000000u) ? (e & 0x7FFFFFFFu) : ~e;
    return __uint_as_float(u);
}
__device__ __forceinline__ float leaky02(float x) { return x > 0.0f ? x : 0.2f * x; }

__global__ void edge_max(const int* __restrict__ src, const int* __restrict__ dst,
                         const float* __restrict__ el, const float* __restrict__ er,
                         unsigned* __restrict__ menc, int E2) {
    int t = blockIdx.x * blockDim.x + threadIdx.x;
    if (t >= E2) return;
    int e = t >> 1, h = t & 1;
    float v = leaky02(el[src[e] * 2 + h] + er[dst[e] * 2 + h]);
    atomicMax(&menc[dst[e] * 2 + h], fenc(v));
}
__global__ void edge_expsum(const int* __restrict__ src, const int* __restrict__ dst,
                            const float* __restrict__ el, const float* __restrict__ er,
                            const unsigned* __restrict__ menc,
                            float* __restrict__ ex, float* __restrict__ s, int E2) {
    int t = blockIdx.x * blockDim.x + threadIdx.x;
    if (t >= E2) return;
    int e = t >> 1, h = t & 1;
    int d2 = dst[e] * 2 + h;
    float v  = leaky02(el[src[e] * 2 + h] + er[d2 - h + h]);  // el[src]+er[dst]
    float m  = fdec(menc[d2]);
    float xv = __expf(v - m);
    ex[t] = xv;
    atomicAdd(&s[d2], xv);
}
__global__ void edge_agg(const int* __restrict__ src, const int* __restrict__ dst,
                         const float* __restrict__ ft, const float* __restrict__ ex,
                         const float* __restrict__ s, float* __restrict__ out, int E32) {
    int t = blockIdx.x * blockDim.x + threadIdx.x;
    if (t >= E32) return;
    int e = t >> 5, f = t & 31, h = f >> 4;
    int d = dst[e];
    float alpha = ex[e * 2 + h] / s[d * 2 + h];
    atomicAdd(&out[(long)d * 32 + f], ft[(long)src[e] * 32 + f] * alpha);
}

// ------------------------------ batch norm ---------------------------------
__global__ void bn_stats(const float* __restrict__ x, int n_rows,
                         float* __restrict__ stats /*64: sum, sumsq*/) {
    __shared__ float ls[64];
    int tid = threadIdx.x;
    if (tid < 64) ls[tid] = 0.0f;
    __syncthreads();
    int col = tid & 31;
    int stride = gridDim.x * blockDim.x;              // multiple of 32
    float sum = 0.0f, sq = 0.0f;
    for (long i = blockIdx.x * blockDim.x + tid; i < (long)n_rows * 32; i += stride) {
        float v = x[i];
        v = v > 0.0f ? v : 0.0f;                      // stats over relu(x)
        sum += v; sq += v * v;
    }
    atomicAdd(&ls[col], sum);
    atomicAdd(&ls[32 + col], sq);
    __syncthreads();
    if (tid < 64) atomicAdd(&stats[tid], ls[tid]);
}
__global__ void bn_finalize(const float* __restrict__ stats, int n_rows,
                            const float* __restrict__ g, const float* __restrict__ be,
                            float* __restrict__ ss /*64: scale, shift*/) {
    int c = threadIdx.x;
    if (c >= 32) return;
    float inv = 1.0f / (float)n_rows;
    float mu  = stats[c] * inv;
    float var = stats[32 + c] * inv - mu * mu;
    float sc  = g[c] * rsqrtf(var + EPSV);
    ss[c]      = sc;
    ss[32 + c] = be[c] - mu * sc;
}
__global__ void bn_apply(float* x, int n, const float* __restrict__ ss) {
    int t = blockIdx.x * blockDim.x + threadIdx.x;
    if (t >= n) return;
    int c = t & 31;
    float v = x[t];
    v = v > 0.0f ? v : 0.0f;
    x[t] = ss[c] * v + ss[32 + c];
}

// ------------------------------- pooling -----------------------------------
__global__ void pool_acc(const float* __restrict__ x, const int* __restrict__ gid,
                         int n_rows, float* __restrict__ sums, float* __restrict__ cnt,
                         int off, int type) {
    int t = blockIdx.x * blockDim.x + threadIdx.x;
    if (t >= n_rows * 32) return;
    int n = t >> 5, c = t & 31;
    int g = gid[n];
    atomicAdd(&sums[(long)g * 96 + off + c], x[t]);
    if (c == 0) atomicAdd(&cnt[g * 3 + type], 1.0f);
}
__global__ void head_out(const float* __restrict__ sums, const float* __restrict__ cnt,
                         const float* __restrict__ W_out, const float* __restrict__ b_out,
                         float* __restrict__ out, int G) {
    int g = blockIdx.x * blockDim.x + threadIdx.x;
    if (g >= G) return;
    float acc = b_out[0];
    for (int type = 0; type < 3; ++type) {
        float c = cnt[g * 3 + type];
        float inv = c > 0.0f ? 1.0f / c : 0.0f;
        for (int j = 0; j < 32; ++j) {
            int k = type * 32 + j;
            acc += (sums[(long)g * 96 + k] * inv) * W_out[k];
        }
    }
    out[g] = acc;
}

// ---------------------------------------------------------------------------
extern "C" void kernel_launch(void* const* d_in, const int* in_sizes, int n_in,
                              void* d_out, int out_size, void* d_ws, size_t ws_size,
                              hipStream_t stream) {
    const float* x_node = (const float*)d_in[0];
    const float* x_edge = (const float*)d_in[1];
    const float* x_cyc  = (const float*)d_in[2];
    const int* src[5], * dst[5];
    for (int r = 0; r < 5; ++r) {
        src[r] = (const int*)d_in[3 + 2 * r];
        dst[r] = (const int*)d_in[4 + 2 * r];
    }
    const int* gid_n = (const int*)d_in[13];
    const int* gid_e = (const int*)d_in[14];
    const int* gid_c = (const int*)d_in[15];
    const float* Wt_n = (const float*)d_in[16]; const float* bt_n = (const float*)d_in[17];
    const float* Wt_e = (const float*)d_in[18]; const float* bt_e = (const float*)d_in[19];
    const float* Wt_c = (const float*)d_in[20]; const float* bt_c = (const float*)d_in[21];
    const float* W1  = (const float*)d_in[22]; const float* al1 = (const float*)d_in[23];
    const float* ar1 = (const float*)d_in[24]; const float* b1  = (const float*)d_in[25];
    const float* W2  = (const float*)d_in[26]; const float* al2 = (const float*)d_in[27];
    const float* ar2 = (const float*)d_in[28]; const float* b2  = (const float*)d_in[29];
    const float* g1  = (const float*)d_in[30]; const float* be1 = (const float*)d_in[31];
    const float* g2  = (const float*)d_in[32]; const float* be2 = (const float*)d_in[33];
    const float* W_out = (const float*)d_in[34]; const float* b_out = (const float*)d_in[35];

    const int NN = in_sizes[0] / 14;
    const int NE = in_sizes[1] / 5;
    const int NC = in_sizes[2] / 8;
    const int G  = out_size;
    int E[5]; for (int r = 0; r < 5; ++r) E[r] = in_sizes[3 + 2 * r];
    int maxN = NN > NE ? NN : NE; if (NC > maxN) maxN = NC;
    int maxE = 0; for (int r = 0; r < 5; ++r) if (E[r] > maxE) maxE = E[r];

    // ---- workspace carve-out (floats) ----
    float* wsf = (float*)d_ws;
    size_t off = 0;
    auto alloc = [&](size_t n) { float* p = wsf + off; off += n; return p; };
    float* hA_n = alloc((size_t)NN * 32);
    float* hA_e = alloc((size_t)NE * 32);
    float* hA_c = alloc((size_t)NC * 32);
    float* hB_n = alloc((size_t)NN * 32);
    float* hB_e = alloc((size_t)NE * 32);
    float* hB_c = alloc((size_t)NC * 32);
    float* ft   = alloc((size_t)maxN * 32);
    float* el   = alloc((size_t)maxN * 2);
    float* er   = alloc((size_t)maxN * 2);
    unsigned* menc = (unsigned*)alloc((size_t)maxN * 2);
    float* ssum = alloc((size_t)maxN * 2);
    float* ex   = alloc((size_t)maxE * 2);
    float* stats = alloc(64);
    float* ssbuf = alloc(64);
    float* psums = alloc((size_t)G * 96);
    float* pcnt  = alloc((size_t)G * 3);
    (void)ws_size; (void)n_in;

    auto grid = [](long n, int b) { return (int)((n + b - 1) / b); };
    auto gemm = [&](const float* A, int M, int K, const float* B, const float* bias,
                    float* C, const float* attn, float* e_out) {
        gemm32_wmma<<<grid(M, GEMM_ROWS), GEMM_THREADS, 0, stream>>>(
            A, M, K, B, bias, C, attn, e_out);
    };

    // ---- input projections (WMMA) ----
    gemm(x_node, NN, 14, Wt_n, bt_n, hA_n, nullptr, nullptr);
    gemm(x_edge, NE, 5,  Wt_e, bt_e, hA_e, nullptr, nullptr);
    gemm(x_cyc,  NC, 8,  Wt_c, bt_c, hA_c, nullptr, nullptr);

    auto run_layer = [&](const float* W, const float* al, const float* ar,
                         const float* b, const float* hn_in, const float* he_in,
                         const float* hc_in, float* hn_out, float* he_out,
                         float* hc_out) {
        // accumulators start at sum of destination-relation biases
        init_bias3<<<grid((long)NN * 32, 256), 256, 0, stream>>>(hn_out, NN, b, b + 64, b + 96);
        init_bias3<<<grid((long)NE * 32, 256), 256, 0, stream>>>(he_out, NE, b + 32, nullptr, nullptr);
        init_bias3<<<grid((long)NC * 32, 256), 256, 0, stream>>>(hc_out, NC, b + 128, nullptr, nullptr);

        const float* hs[5] = { hn_in, hn_in, he_in, hc_in, hn_in };
        const float* hd[5] = { hn_in, he_in, hn_in, hn_in, hc_in };
        int Ns[5] = { NN, NN, NE, NC, NN };
        int Nd[5] = { NN, NE, NN, NN, NC };
        float* acc[5] = { hn_out, he_out, hn_out, hn_out, hc_out };

        for (int r = 0; r < 5; ++r) {
            const float* Wr = W + r * 1024;
            // source transform: store ft, fused el ; dest transform: er only
            gemm(hs[r], Ns[r], 32, Wr, nullptr, ft, al + r * 32, el);
            gemm(hd[r], Nd[r], 32, Wr, nullptr, nullptr, ar + r * 32, er);
            // segment softmax over dst
            fill_u32<<<grid((long)Nd[r] * 2, 256), 256, 0, stream>>>(menc, Nd[r] * 2, 0x007FFFFFu);
            fill_f32<<<grid((long)Nd[r] * 2, 256), 256, 0, stream>>>(ssum, Nd[r] * 2, 0.0f);
            edge_max<<<grid((long)E[r] * 2, 256), 256, 0, stream>>>(src[r], dst[r], el, er, menc, E[r] * 2);
            edge_expsum<<<grid((long)E[r] * 2, 256), 256, 0, stream>>>(src[r], dst[r], el, er, menc, ex, ssum, E[r] * 2);
            edge_agg<<<grid((long)E[r] * 32, 256), 256, 0, stream>>>(src[r], dst[r], ft, ex, ssum, acc[r], E[r] * 32);
        }
        // HeteroGraphConv aggregate='mean' over 3 node-destination relations
        scale_inplace<<<grid((long)NN * 32, 256), 256, 0, stream>>>(hn_out, NN * 32, 1.0f / 3.0f);
    };

    auto run_bn = [&](float* x, int n_rows, const float* g, const float* be) {
        fill_f32<<<1, 64, 0, stream>>>(stats, 64, 0.0f);
        bn_stats<<<512, 256, 0, stream>>>(x, n_rows, stats);
        bn_finalize<<<1, 32, 0, stream>>>(stats, n_rows, g, be, ssbuf);
        bn_apply<<<grid((long)n_rows * 32, 256), 256, 0, stream>>>(x, n_rows * 32, ssbuf);
    };

    // ---- layer 1 ----
    run_layer(W1, al1, ar1, b1, hA_n, hA_e, hA_c, hB_n, hB_e, hB_c);
    run_bn(hB_n, NN, g1, be1);
    run_bn(hB_e, NE, g1, be1);
    run_bn(hB_c, NC, g1, be1);
    // ---- layer 2 ----
    run_layer(W2, al2, ar2, b2, hB_n, hB_e, hB_c, hA_n, hA_e, hA_c);
    run_bn(hA_n, NN, g2, be2);
    run_bn(hA_e, NE, g2, be2);
    run_bn(hA_c, NC, g2, be2);

    // ---- pooling + head ----
    fill_f32<<<grid((long)G * 96, 256), 256, 0, stream>>>(psums, G * 96, 0.0f);
    fill_f32<<<grid((long)G * 3, 256), 256, 0, stream>>>(pcnt, G * 3, 0.0f);
    pool_acc<<<grid((long)NN * 32, 256), 256, 0, stream>>>(hA_n, gid_n, NN, psums, pcnt, 0, 0);
    pool_acc<<<grid((long)NE * 32, 256), 256, 0, stream>>>(hA_e, gid_e, NE, psums, pcnt, 32, 1);
    pool_acc<<<grid((long)NC * 32, 256), 256, 0, stream>>>(hA_c, gid_c, NC, psums, pcnt, 64, 2);
    head_out<<<grid(G, 256), 256, 0, stream>>>(psums, pcnt, W_out, b_out, (float*)d_out, G);
}